// QuantumDecoder_65481071407499
// MI455X (gfx1250) — compile-verified
//
#include <hip/hip_runtime.h>

// ---------------------------------------------------------------------------
// 20-qubit statevector "QuantumDecoder" for MI455X (gfx1250, wave32).
//
// Fusions (exact):
//  * encoding RY + rep0 Rot  -> product state (2-vectors per wire/sample)
//  * CNOT chains             -> basis permutations; inverse = Gray map
//                               idx -> idx ^ (idx>>1); rep1 chain folded into
//                               the Z-expectation signs (prefix parity).
//  * rep1 Rot layer          -> 3-wire fused gates = 16x16 real matrices,
//                               applied with v_wmma_f32_16x16x4_f32.
// Data movement (CDNA5 paths):
//  * k_stage1 writeback: global_store_async_from_lds_b128 (+ s_wait_asynccnt)
//  * k_stage2 gather:    TDM tensor_load_to_lds, one 2D strided tile per
//                        block (+ s_wait_tensorcnt)
// Traffic: 64MB write (stage1, incl. wires 5..19) + 64MB read (stage2).
// ---------------------------------------------------------------------------

typedef float v2f __attribute__((ext_vector_type(2)));
typedef float v8f __attribute__((ext_vector_type(8)));
typedef unsigned int u32x4 __attribute__((ext_vector_type(4)));
typedef int i32x4 __attribute__((ext_vector_type(4)));
typedef int i32x8 __attribute__((ext_vector_type(8)));

#define NQ    20
#define DIM   (1u << 20)
#define BATCH 8

struct c32 { float x, y; };
__device__ __forceinline__ c32 cmul(c32 a, c32 b) {
    return { a.x * b.x - a.y * b.y, a.x * b.y + a.y * b.x };
}
__device__ __forceinline__ c32 cadd(c32 a, c32 b) { return { a.x + b.x, a.y + b.y }; }

// ---------------------------------------------------------------------------
// Apply a fused 3-adjacent-bit gate (16x16 real matrix M, layout [[Re,-Im],[Im,Re]])
// to an LDS tile of 2^15 complex amplitudes, group bits at positions p..p+2.
// Column vector = [re(g=0..7); im(g=0..7)].  D = M * X via 4x WMMA f32 16x16x4.
//
// WMMA f32 operand layout (ISA 7.12.2):
//   A 16x4 : lane L holds row (L&15), K = {kb, kb+1}, kb = (L>>4)*2
//   B 4x16 : lane L holds col (L&15), K rows {kb, kb+1}
//   C/D    : vgpr d, lane L -> row d + (L>>4)*8, col (L&15)
// ---------------------------------------------------------------------------
__device__ __forceinline__ void apply_group16(float2* tile, const float* __restrict__ M,
                                              int p, int lane, int wave, int nwaves)
{
    const int row = lane & 15;
    const int kb  = (lane >> 4) << 1;            // 0 or 2
    v2f a0 = { M[row * 16 +  0 + kb], M[row * 16 +  1 + kb] };
    v2f a1 = { M[row * 16 +  4 + kb], M[row * 16 +  5 + kb] };
    v2f a2 = { M[row * 16 +  8 + kb], M[row * 16 +  9 + kb] };
    v2f a3 = { M[row * 16 + 12 + kb], M[row * 16 + 13 + kb] };
    const int pm = (1 << p) - 1;
    float* t = (float*)tile;

    // 2^12 columns, 16 per WMMA, split over waves (uniform trip count per wave).
    for (int s = wave; s < 256; s += nwaves) {
        int col = (s << 4) | row;                            // this lane's column
        int cb  = ((col >> p) << (p + 3)) | (col & pm);      // insert 3 zero bits at p
        v8f acc = { 0.f, 0.f, 0.f, 0.f, 0.f, 0.f, 0.f, 0.f };
#pragma unroll
        for (int k = 0; k < 4; ++k) {
            int r0 = 4 * k + kb, r1 = r0 + 1;
            v2f b;
            b.x = t[((cb | ((r0 & 7) << p)) << 1) | (r0 >> 3)];
            b.y = t[((cb | ((r1 & 7) << p)) << 1) | (r1 >> 3)];
            v2f a = (k == 0) ? a0 : (k == 1) ? a1 : (k == 2) ? a2 : a3;
            acc = __builtin_amdgcn_wmma_f32_16x16x4_f32(
                false, a, false, b, (short)0, acc, false, false);
        }
        int rbase = (lane >> 4) << 3;
#pragma unroll
        for (int d = 0; d < 8; ++d) {
            int r = rbase + d;
            t[((cb | ((r & 7) << p)) << 1) | (r >> 3)] = acc[d];
        }
    }
}

// ---------------------------------------------------------------------------
// Kernel 1: tables.  Per-sample 2-vectors v[b][w] = Rot_rep0(w) * RY(latent)|0>,
// rep1 per-wire 2x2 U's, and six 16x16 real fused-group matrices:
//   gi=0..4 -> wires (5+3gi, 6+3gi, 7+3gi); gi=5 -> wires (0,1,2).
// Also zeroes d_out (deterministic accumulation target).
// ---------------------------------------------------------------------------
__global__ __launch_bounds__(256)
void k_prep(const float* __restrict__ latent, const float* __restrict__ weight,
            float2* __restrict__ vtab, float2* __restrict__ utab,
            float* __restrict__ mtab, float* __restrict__ out)
{
    __shared__ c32 Us[NQ][2][2];
    int tid = threadIdx.x;
    if (tid < BATCH * NQ) out[tid] = 0.f;

    if (tid < NQ) {                              // rep-1 Rot matrices
        int w = tid;
        float phi = weight[60 + 3 * w], th = weight[60 + 3 * w + 1], om = weight[60 + 3 * w + 2];
        float ct = cosf(0.5f * th), st = sinf(0.5f * th);
        float al = 0.5f * (phi + om), be = 0.5f * (phi - om);
        c32 ep = { cosf(al), -sinf(al) }, em = { cosf(be), -sinf(be) };
        c32 u00 = {  ep.x * ct,  ep.y * ct };
        c32 u01 = { -em.x * st,  em.y * st };    // -conj(em)*s
        c32 u10 = {  em.x * st,  em.y * st };
        c32 u11 = {  ep.x * ct, -ep.y * ct };    //  conj(ep)*c
        Us[w][0][0] = u00; Us[w][0][1] = u01; Us[w][1][0] = u10; Us[w][1][1] = u11;
        utab[w * 4 + 0] = { u00.x, u00.y }; utab[w * 4 + 1] = { u01.x, u01.y };
        utab[w * 4 + 2] = { u10.x, u10.y }; utab[w * 4 + 3] = { u11.x, u11.y };
    }
    if (tid < BATCH * NQ) {                      // v = Rot_rep0 * RY(latent) |0>
        int b = tid / NQ, w = tid % NQ;
        float phi = weight[3 * w], th = weight[3 * w + 1], om = weight[3 * w + 2];
        float ct = cosf(0.5f * th), st = sinf(0.5f * th);
        float al = 0.5f * (phi + om), be = 0.5f * (phi - om);
        c32 ep = { cosf(al), -sinf(al) }, em = { cosf(be), -sinf(be) };
        c32 u00 = {  ep.x * ct,  ep.y * ct };
        c32 u01 = { -em.x * st,  em.y * st };
        c32 u10 = {  em.x * st,  em.y * st };
        c32 u11 = {  ep.x * ct, -ep.y * ct };
        float L = latent[b * NQ + w];
        float ca = cosf(0.5f * L), sa = sinf(0.5f * L);
        vtab[(b * NQ + w) * 2 + 0] = { u00.x * ca + u01.x * sa, u00.y * ca + u01.y * sa };
        vtab[(b * NQ + w) * 2 + 1] = { u10.x * ca + u11.x * sa, u10.y * ca + u11.y * sa };
    }
    __syncthreads();
    for (int e = tid; e < 6 * 256; e += 256) {   // real-rep Kronecker matrices
        int gi = e >> 8, rc = e & 255, r = rc >> 4, c = rc & 15;
        int w0 = (gi < 5) ? (5 + 3 * gi) : 0;
        int gr = r & 7, gc = c & 7;
        c32 g = cmul(cmul(Us[w0][(gr >> 2) & 1][(gc >> 2) & 1],
                          Us[w0 + 1][(gr >> 1) & 1][(gc >> 1) & 1]),
                     Us[w0 + 2][gr & 1][gc & 1]);
        float val = ((r < 8) == (c < 8)) ? g.x : ((r >= 8) ? g.y : -g.y);
        mtab[gi * 256 + r * 16 + c] = val;
    }
}

// ---------------------------------------------------------------------------
// Kernel 2: build state after rep0 (product state composed with CNOT-chain
// permutation, i.e. Gray-coordinate outer product), then apply rep1 Rot on
// wires 5..19 as 5 WMMA group-gates.  Writeback via async LDS->global B128
// stores (ASYNCcnt path).  Block = (sample, top-5-bits); tile = 2^15
// contiguous amplitudes (256KB dynamic LDS).
// ---------------------------------------------------------------------------
__global__ __launch_bounds__(256)
void k_stage1(const float2* __restrict__ vtab, const float* __restrict__ mtab,
              float2* __restrict__ state)
{
    extern __shared__ float2 tile[];             // 2^15 complex (dynamic, 256KB)
    __shared__ float2 tbl[256];                  // products of factors j=7..14
    int tid = threadIdx.x;
    int smp = blockIdx.x >> 5, hi5 = blockIdx.x & 31;
    const float2* v = vtab + smp * (NQ * 2);     // v[w*2 + bit], t_j = v[19-j]

    {   // tbl[m] = prod_{j=7..14} t_j[bit_{j-7}(m)]
        c32 pr = { 1.f, 0.f };
        for (int j = 7; j <= 14; ++j) {
            float2 tv = v[(19 - j) * 2 + ((tid >> (j - 7)) & 1)];
            pr = cmul(pr, { tv.x, tv.y });
        }
        tbl[tid] = { pr.x, pr.y };
    }
    c32 tc = { 1.f, 0.f };                       // factors j=15..19 (block) and 0..6 (thread)
    {
        int xh = hi5 ^ (hi5 >> 1);
        for (int j = 15; j <= 19; ++j) {
            float2 tv = v[(19 - j) * 2 + ((xh >> (j - 15)) & 1)];
            tc = cmul(tc, { tv.x, tv.y });
        }
        int g7 = tid ^ (tid >> 1);
        for (int j = 0; j <= 6; ++j) {
            float2 tv = v[(19 - j) * 2 + ((g7 >> j) & 1)];
            tc = cmul(tc, { tv.x, tv.y });
        }
    }
    __syncthreads();
    unsigned idxhi = (unsigned)hi5 << 15;
    for (int i = 0; i < 128; ++i) {              // 1 cmul per amplitude
        unsigned lo  = ((unsigned)i << 8) | tid;
        unsigned idx = idxhi | lo;
        unsigned x   = idx ^ (idx >> 1);         // Gray map = inverse CNOT chain
        float2 tv = tbl[(x >> 7) & 255];
        c32 amp = cmul(tc, { tv.x, tv.y });
        tile[lo] = { amp.x, amp.y };
    }
    __syncthreads();

    int lane = tid & 31, wave = tid >> 5;
    for (int gi = 0; gi < 5; ++gi) {             // wires (5,6,7)..(17,18,19) -> p=12..0
        apply_group16(tile, mtab + gi * 256, 12 - 3 * gi, lane, wave, 8);
        __syncthreads();
    }

    // Async writeback: 2^14 x 16B straight from LDS to HBM (never touches VGPRs).
    {
        unsigned ldsb = (unsigned)(unsigned long long)(void*)tile;
        unsigned long long gd = (unsigned long long)(void*)
            (state + ((size_t)smp << 20) + ((size_t)hi5 << 15));
        for (int i = tid; i < (1 << 14); i += 256) {
            unsigned          l = ldsb + ((unsigned)i << 4);
            unsigned long long g = gd + ((unsigned long long)i << 4);
            asm volatile("global_store_async_from_lds_b128 %0, %1, off"
                         :: "v"(g), "v"(l) : "memory");
        }
        asm volatile("s_wait_asynccnt 0x0" ::: "memory");
    }
}

// ---------------------------------------------------------------------------
// Kernel 3: TDM-gather a 2D strided tile holding all 32 combos of the top 5
// bits (mid bits fixed) straight into LDS, apply wires {0,1,2} via WMMA and
// wires {3,4} via VALU, then accumulate <Z_w> with the rep1 CNOT chain folded
// into prefix-parity signs.  Read-only from HBM (64 MB) + tiny atomics.
// ---------------------------------------------------------------------------
__global__ __launch_bounds__(256)
void k_stage2(const float2* __restrict__ state, const float2* __restrict__ utab,
              const float* __restrict__ mtab, float* __restrict__ out)
{
    extern __shared__ float2 tile[];             // 2^15 complex, addr = (hi5<<10)|lo10
    __shared__ float red[NQ];
    int tid = threadIdx.x;
    int smp = blockIdx.x >> 5, mid = blockIdx.x & 31;

    // One Tensor-Data-Mover descriptor per block: 2D tile, 32 rows x 1024
    // complex64 (8KB), row stride 2^15 elements.  TDM ignores EXEC, so only
    // wave 0 issues it; everyone else parks on the barrier.
    if (tid < 32) {
        unsigned           ldsb = (unsigned)(unsigned long long)(void*)tile;
        unsigned long long ga   = (unsigned long long)(const void*)
            (state + ((size_t)smp << 20) + ((size_t)mid << 10));
        // D# group0: count=1 | lds_addr | global_addr | type=2 (bits 127:126)
        u32x4 g0 = { 1u, ldsb, (unsigned)ga, (unsigned)(ga >> 32) | 0x80000000u };
        // D# group1: data_size=8B; tensor_dim0=1024; tensor_dim1=32;
        //            tile_dim0=1024; tile_dim1=32; tensor_dim0_stride=32768.
        i32x8 g1 = { 0x30000,
                     (int)(1024u << 16),        // tensor_dim0[15:0] << 16
                     (int)(32u << 16),          // tensor_dim1[15:0] << 16
                     (int)(1024u << 16),        // tile_dim0 << 16
                     32,                        // tile_dim1
                     32768,                     // tensor_dim0_stride lo32
                     0, 0 };
        i32x4 gz4 = { 0, 0, 0, 0 };
        i32x8 gz8 = { 0, 0, 0, 0, 0, 0, 0, 0 };
        __builtin_amdgcn_tensor_load_to_lds(g0, g1, gz4, gz4, gz8, 0);
        __builtin_amdgcn_s_wait_tensorcnt(0);
    }
    __syncthreads();

    int lane = tid & 31, wave = tid >> 5;
    apply_group16(tile, mtab + 5 * 256, 12, lane, wave, 8);   // wires 0,1,2 (tile bits 14..12)
    __syncthreads();

    for (int w = 3; w <= 4; ++w) {               // wires 3,4 -> tile bits 11,10
        int q = 14 - w;
        c32 u00 = { utab[w * 4 + 0].x, utab[w * 4 + 0].y };
        c32 u01 = { utab[w * 4 + 1].x, utab[w * 4 + 1].y };
        c32 u10 = { utab[w * 4 + 2].x, utab[w * 4 + 2].y };
        c32 u11 = { utab[w * 4 + 3].x, utab[w * 4 + 3].y };
        for (int i = tid; i < (1 << 14); i += 256) {
            int a0 = ((i >> q) << (q + 1)) | (i & ((1 << q) - 1));
            int a1 = a0 | (1 << q);
            c32 x0 = { tile[a0].x, tile[a0].y };
            c32 x1 = { tile[a1].x, tile[a1].y };
            c32 y0 = cadd(cmul(u00, x0), cmul(u01, x1));
            c32 y1 = cadd(cmul(u10, x0), cmul(u11, x1));
            tile[a0] = { y0.x, y0.y };
            tile[a1] = { y1.x, y1.y };
        }
        __syncthreads();
    }

    float acc[NQ];
#pragma unroll
    for (int w = 0; w < NQ; ++w) acc[w] = 0.f;
    unsigned idmid = (unsigned)mid << 10;
    for (int i = 0; i < 128; ++i) {
        unsigned a   = ((unsigned)i << 8) | tid;
        unsigned idx = ((a >> 10) << 15) | idmid | (a & 1023);
        unsigned z = idx;                        // z_j = parity of idx bits >= j
        z ^= z >> 1; z ^= z >> 2; z ^= z >> 4; z ^= z >> 8; z ^= z >> 16;
        float2 am = tile[a];
        float pr = am.x * am.x + am.y * am.y;
#pragma unroll
        for (int w = 0; w < NQ; ++w)
            acc[w] += ((z >> (19 - w)) & 1) ? -pr : pr;
    }
    if (tid < NQ) red[tid] = 0.f;
    __syncthreads();
#pragma unroll
    for (int w = 0; w < NQ; ++w) atomicAdd(&red[w], acc[w]);
    __syncthreads();
    if (tid < NQ) atomicAdd(&out[smp * NQ + tid], red[tid]);
}

// ---------------------------------------------------------------------------
extern "C" void kernel_launch(void* const* d_in, const int* in_sizes, int n_in,
                              void* d_out, int out_size, void* d_ws, size_t ws_size,
                              hipStream_t stream) {
    (void)in_sizes; (void)n_in; (void)out_size; (void)ws_size;
    const float* latent = (const float*)d_in[0];   // (8,20) f32
    const float* weight = (const float*)d_in[1];   // (120,) f32
    float*       out    = (float*)d_out;           // (8,20) f32

    float2* state = (float2*)d_ws;                             // 8 * 2^20 complex (64MB)
    float2* vtab  = state + (size_t)BATCH * DIM;               // 8*20*2 complex
    float2* utab  = vtab + BATCH * NQ * 2;                     // 20*4 complex
    float*  mtab  = (float*)(utab + NQ * 4);                   // 6 * 256 f32

    k_prep  <<<1,   256, 0, stream>>>(latent, weight, vtab, utab, mtab, out);
    k_stage1<<<256, 256, (1u << 15) * sizeof(float2), stream>>>(vtab, mtab, state);
    k_stage2<<<256, 256, (1u << 15) * sizeof(float2), stream>>>(state, utab, mtab, out);
}